// VertexMatcher_60017872994577
// MI455X (gfx1250) — compile-verified
//
#include <hip/hip_runtime.h>

// Nearest-neighbor correspondence via WMMA:
//   score(m,n) = ||b_n||^2 - 2 a_m . b_n      (argmin-equivalent to full sq dist)
//   A'[m] = (-2ax,-2ay,-2az, 1)   B'[n] = (bx,by,bz, ||b||^2)
//   score tile = A'(16x4) x B'(4x16)  ->  V_WMMA_F32_16X16X4_F32, K=4 exact fit.

#define N_MHR   20000
#define N_SMPLX 10475
#define N_PAD   10480   // 655 * 16 columns (pad scores = +inf)
#define NT_COL  655
#define NT_ROW  1250    // 20000 / 16 rows, exact

typedef float v2f __attribute__((ext_vector_type(2)));
typedef float v8f __attribute__((ext_vector_type(8)));

__global__ __launch_bounds__(256) void vm_prep(const float* __restrict__ mhr,
                                               const float* __restrict__ smplx,
                                               float4* __restrict__ A4,
                                               float4* __restrict__ B4) {
  int i = blockIdx.x * blockDim.x + threadIdx.x;
  if (i < N_MHR) {
    float x = mhr[3 * i + 0], y = mhr[3 * i + 1], z = mhr[3 * i + 2];
    A4[i] = make_float4(-2.0f * x, -2.0f * y, -2.0f * z, 1.0f);
  }
  if (i < N_PAD) {
    if (i < N_SMPLX) {
      float x = smplx[3 * i + 0], y = smplx[3 * i + 1], z = smplx[3 * i + 2];
      B4[i] = make_float4(x, y, z, x * x + y * y + z * z);
    } else {
      B4[i] = make_float4(0.0f, 0.0f, 0.0f, __builtin_inff());  // never argmin
    }
  }
}

__global__ __launch_bounds__(256) void vm_match(const float4* __restrict__ A4,
                                                const float4* __restrict__ B4,
                                                int* __restrict__ out) {
  // Stage the entire padded B' table in LDS: 10480 * 16B = 167,680 B
  // (CDNA5 WGP allows up to 320KB per workgroup).
  extern __shared__ float4 Bs4[];
  float2* Bs2 = (float2*)Bs4;

  for (int t = threadIdx.x; t < N_PAD; t += 256) Bs4[t] = B4[t];
  __syncthreads();

  const float2* A2 = (const float2*)A4;
  const int lane = threadIdx.x & 31;
  const int wave = threadIdx.x >> 5;
  const int nlo  = lane & 15;   // matrix column within tile (C/D: lane = N)
  const int kp   = lane >> 4;   // K half: lanes 0-15 -> K={0,1}, 16-31 -> K={2,3}

  // One 16-row tile per wave per iteration; guard is wave-uniform so EXEC is
  // all-ones at every WMMA (ISA requirement).
  for (int rt = blockIdx.x * 8 + wave; rt < NT_ROW; rt += gridDim.x * 8) {
    const int m = rt * 16 + nlo;                 // A fragment: lane = M row
    const float2 a2 = A2[2 * m + kp];            // K pair for this half-wave
    const v2f af = {a2.x, a2.y};

    float vmin[8];
    int   imin[8];
#pragma unroll
    for (int r = 0; r < 8; ++r) { vmin[r] = __builtin_inff(); imin[r] = 0; }

    // Unroll x2 so the compiler can pipeline ds_load_b64 of the next B tile
    // across the current WMMA + argmin bookkeeping (NT_COL = 655 is odd, but
    // clang peels the remainder).
#pragma unroll 2
    for (int ct = 0; ct < NT_COL; ++ct) {
      const int n = ct * 16 + nlo;               // B fragment: lane = N column
      const float2 b2 = Bs2[2 * n + kp];         // ds_load_b64 from staged B'
      const v2f bf = {b2.x, b2.y};
      v8f c = {0.f, 0.f, 0.f, 0.f, 0.f, 0.f, 0.f, 0.f};
      // (neg_a, A, neg_b, B, c_mod, C, reuse_a, reuse_b)
      c = __builtin_amdgcn_wmma_f32_16x16x4_f32(false, af, false, bf,
                                                (short)0, c, false, false);
#pragma unroll
      for (int r = 0; r < 8; ++r) {              // running per-lane argmin
        float cv = c[r];
        if (cv < vmin[r]) { vmin[r] = cv; imin[r] = n; }
      }
    }

    // Reduce argmin across the 16 lanes of each wave half (xor masks < 16
    // stay inside each half under wave32). Tie-break: smaller column index,
    // matching jnp.argmin first-occurrence semantics. Non-short-circuit
    // logic keeps this branchless (v_cmp + v_cndmask, no exec juggling).
#pragma unroll
    for (int r = 0; r < 8; ++r) {
      float v = vmin[r];
      int   idx = imin[r];
#pragma unroll
      for (int mask = 8; mask >= 1; mask >>= 1) {
        float vo = __shfl_xor(v, mask, 32);
        int   io = __shfl_xor(idx, mask, 32);
        const bool better = (vo < v) | ((vo == v) & (io < idx));
        idx = better ? io : idx;
        v   = better ? vo : v;
      }
      if (nlo == 0) {
        // C/D layout: VGPR r -> row r (lanes 0-15) or row r+8 (lanes 16-31)
        out[rt * 16 + r + (kp << 3)] = idx;
      }
    }
  }
}

extern "C" void kernel_launch(void* const* d_in, const int* in_sizes, int n_in,
                              void* d_out, int out_size, void* d_ws, size_t ws_size,
                              hipStream_t stream) {
  const float* mhr   = (const float*)d_in[0];   // [20000,3] f32
  const float* smplx = (const float*)d_in[1];   // [10475,3] f32
  int* out = (int*)d_out;                       // [20000] int32 argmin indices

  char* ws = (char*)d_ws;
  float4* A4 = (float4*)ws;                         // 20000*16 = 320000 B
  float4* B4 = (float4*)(ws + (size_t)N_MHR * 16);  // 10480*16 = 167680 B

  vm_prep<<<(N_MHR + 255) / 256, 256, 0, stream>>>(mhr, smplx, A4, B4);

  const size_t lds_bytes = (size_t)N_PAD * sizeof(float4);  // 167,680 B
  vm_match<<<64, 256, lds_bytes, stream>>>(A4, B4, out);
}